// edge_10969346474089
// MI455X (gfx1250) — compile-verified
//
#include <hip/hip_runtime.h>
#include <hip/hip_bf16.h>

// Edge-detect pipeline: separable 5-tap Gaussian blur -> Sobel gx/gy ->
// sqrt(sum_c gx^2+gy^2) -> divide by global max.  [32,3,512,512] f32 in,
// [32,1,512,512] f32 out.  Bandwidth-bound: use CDNA5
// async global->LDS (ASYNCcnt) with 16B-aligned b128 vectors for the tile
// fill; all three conv stages run out of the 320KB/WGP LDS.

#define IMG_W   512
#define IMG_H   512
#define TILE_W  128
#define TILE_H  16
#define HALO_Y  3
#define HALO_XL 4                   // left x-halo padded to 4 for 16B alignment
#define RAW_W   136                 // x in [-4, 131]; need [-3, 130]
#define RAW_H   (TILE_H + 2*HALO_Y) // 22 ; y in [-3, 18]
#define RAW_S   136                 // row stride (floats); 544B, 16B aligned
#define RAW_V   (RAW_W / 4)         // 34 float4 per row
#define HB_W    (TILE_W + 2)        // 130  (out x in [-1, 128])
#define HB_H    RAW_H               // 22
#define HB_S    132
#define VB_W    HB_W                // 130
#define VB_H    (TILE_H + 2)        // 18   (out y in [-1, 16])
#define VB_S    132
#define RAW_BASE 0
#define HB_BASE  (RAW_S * RAW_H)            // 2992
#define VB_BASE  (HB_BASE + HB_S * HB_H)    // 5896
#define SMEM_TOT (VB_BASE + VB_S * VB_H)    // 8272 floats ~= 33 KB LDS

__global__ __launch_bounds__(256)
void edge_pass1(const float* __restrict__ img, const float* __restrict__ gauss,
                const float* __restrict__ sobel, float* __restrict__ out,
                unsigned* __restrict__ maxbits)
{
    __shared__ float smem[SMEM_TOT];
    const int tid = threadIdx.x;
    const int ox  = blockIdx.x * TILE_W;
    const int oy  = blockIdx.y * TILE_H;
    const int b   = blockIdx.z;

    // Filter taps from the input tensors (not hardcoded).
    const float w0 = gauss[0], w1 = gauss[1], w2 = gauss[2], w3 = gauss[3], w4 = gauss[4];
    const float s00 = sobel[0], s01 = sobel[1], s02 = sobel[2];
    const float s10 = sobel[3], s11 = sobel[4], s12 = sobel[5];
    const float s20 = sobel[6], s21 = sobel[7], s22 = sobel[8];

    const int tx = tid & (TILE_W - 1);  // 0..127
    const int ty = tid >> 7;            // 0..1 ; thread covers rows ty, ty+2, ..., ty+14

    float acc[8];
#pragma unroll
    for (int k = 0; k < 8; ++k) acc[k] = 0.0f;

    for (int c = 0; c < 3; ++c) {
        const float* src = img + ((size_t)(b * 3 + c) << 18);  // * 512*512

        // ---- Phase A: async b128 copy of raw tile (zero SAME-padding) into LDS.
        // Vector start gx = ox-4+4m is 4-aligned and IMG_W%4==0, so each float4
        // is entirely in-bounds or entirely out-of-bounds.
        for (int idx = tid; idx < RAW_V * RAW_H; idx += 256) {
            const unsigned u  = (unsigned)idx;
            const int ly = (int)(u / RAW_V);
            const int m  = (int)(u - (unsigned)ly * RAW_V);
            const int gx = ox - HALO_XL + 4 * m;     // 4-aligned
            const int gy = oy - HALO_Y + ly;
            const int le = RAW_BASE + ly * RAW_S + 4 * m;  // 16B-aligned element
            if ((unsigned)gx < (unsigned)IMG_W && (unsigned)gy < (unsigned)IMG_H) {
                const unsigned ldsoff = (unsigned)le * 4u;                 // LDS bytes
                const unsigned goff   = ((unsigned)((gy << 9) + gx)) << 2; // global bytes
                asm volatile("global_load_async_to_lds_b128 %0, %1, %2"
                             :: "v"(ldsoff), "v"(goff), "s"(src) : "memory");
            } else {
                *(float4*)&smem[le] = make_float4(0.f, 0.f, 0.f, 0.f);  // zero pad
            }
        }
        asm volatile("s_wait_asynccnt 0" ::: "memory");
        __syncthreads();

        // ---- Phase B: horizontal 5-tap blur.
        // hb[y][i] (out x = i-1) needs raw lx = i+1 .. i+5 (left halo is 4).
        for (int idx = tid; idx < HB_W * HB_H; idx += 256) {
            const unsigned u = (unsigned)idx;
            const int y = (int)(u / HB_W);
            const int x = (int)(u - (unsigned)y * HB_W);
            const float* r = &smem[RAW_BASE + y * RAW_S + x + 1];
            smem[HB_BASE + y * HB_S + x] =
                w0 * r[0] + w1 * r[1] + w2 * r[2] + w3 * r[3] + w4 * r[4];
        }
        __syncthreads();

        // ---- Phase C: vertical 5-tap blur  vb[j][i] = sum_k w[k]*hb[j+k][i]
        for (int idx = tid; idx < VB_W * VB_H; idx += 256) {
            const unsigned u = (unsigned)idx;
            const int y = (int)(u / VB_W);
            const int x = (int)(u - (unsigned)y * VB_W);
            const float* h = &smem[HB_BASE + y * HB_S + x];
            smem[VB_BASE + y * VB_S + x] =
                w0 * h[0] + w1 * h[HB_S] + w2 * h[2 * HB_S] +
                w3 * h[3 * HB_S] + w4 * h[4 * HB_S];
        }
        __syncthreads();

        // ---- Phase D: Sobel gx/gy (cross-correlation, gy = transpose), accumulate
#pragma unroll
        for (int k = 0; k < 8; ++k) {
            const int y = ty + 2 * k;      // local output row
            const int i = tx + 1;          // vb x index
            const int j = y + 1;           // vb y index
            const float* v = &smem[VB_BASE + (j - 1) * VB_S + (i - 1)];
            const float a00 = v[0],          a01 = v[1],            a02 = v[2];
            const float a10 = v[VB_S],       a11 = v[VB_S + 1],     a12 = v[VB_S + 2];
            const float a20 = v[2 * VB_S],   a21 = v[2 * VB_S + 1], a22 = v[2 * VB_S + 2];
            const float gx = s00 * a00 + s01 * a01 + s02 * a02
                           + s10 * a10 + s11 * a11 + s12 * a12
                           + s20 * a20 + s21 * a21 + s22 * a22;
            const float gy = s00 * a00 + s10 * a01 + s20 * a02
                           + s01 * a10 + s11 * a11 + s21 * a12
                           + s02 * a20 + s12 * a21 + s22 * a22;
            acc[k] += gx * gx + gy * gy;
        }
        __syncthreads();  // protect raw/hb/vb before next channel overwrites
    }

    // ---- magnitude, store, block-level max
    float m = 0.0f;
#pragma unroll
    for (int k = 0; k < 8; ++k) {
        const int y = ty + 2 * k;
        const float mag = sqrtf(acc[k]);
        out[((size_t)b << 18) + (size_t)(oy + y) * IMG_W + (ox + tx)] = mag;
        m = fmaxf(m, mag);
    }
#pragma unroll
    for (int off = 16; off > 0; off >>= 1)
        m = fmaxf(m, __shfl_xor(m, off));         // wave32 reduction
    const int wid  = tid >> 5;
    const int lane = tid & 31;
    if (lane == 0) smem[wid] = m;                  // reuse raw region (idle now)
    __syncthreads();
    if (tid == 0) {
        float bm = smem[0];
        for (int wv = 1; wv < 8; ++wv) bm = fmaxf(bm, smem[wv]);
        atomicMax(maxbits, __float_as_uint(bm));   // all vals >= 0: bit max == float max
    }
}

__global__ __launch_bounds__(256)
void edge_pass2(float* __restrict__ out, const unsigned* __restrict__ maxbits, int n4)
{
    const float inv = 1.0f / __uint_as_float(*maxbits);
    float4* o = (float4*)out;
    const int stride = gridDim.x * blockDim.x;
    for (int i = blockIdx.x * blockDim.x + threadIdx.x; i < n4; i += stride) {
        float4 v = o[i];
        v.x *= inv; v.y *= inv; v.z *= inv; v.w *= inv;
        o[i] = v;
    }
}

extern "C" void kernel_launch(void* const* d_in, const int* in_sizes, int n_in,
                              void* d_out, int out_size, void* d_ws, size_t ws_size,
                              hipStream_t stream) {
    const float* img   = (const float*)d_in[0];   // [32,3,512,512]
    const float* gauss = (const float*)d_in[1];   // [5]
    const float* sobel = (const float*)d_in[2];   // [3,3]
    float* out = (float*)d_out;                   // [32,1,512,512]
    unsigned* maxbits = (unsigned*)d_ws;

    // reset the global-max cell each call (graph-capture safe)
    hipMemsetAsync(d_ws, 0, sizeof(unsigned), stream);

    dim3 grid(IMG_W / TILE_W, IMG_H / TILE_H, 32);  // (4, 32, 32)
    edge_pass1<<<grid, 256, 0, stream>>>(img, gauss, sobel, out, maxbits);

    const int n4 = out_size / 4;                    // 2,097,152 float4s
    edge_pass2<<<4096, 256, 0, stream>>>(out, maxbits, n4);
}